// CorefGRU_54546084659872
// MI455X (gfx1250) — compile-verified
//
#include <hip/hip_runtime.h>
#include <math.h>

#define R_   4
#define RD_  64
#define D_   256
#define C_   64
#define B_   32
#define T_   512
#define OUT_ 256   // R_*RD_

typedef __attribute__((ext_vector_type(2))) float v2f;
typedef __attribute__((ext_vector_type(4))) float v4f;
typedef __attribute__((ext_vector_type(8))) float v8f;

// ---------------------------------------------------------------------------
// Phase 1a: XW = Xflat (16384 x 256) @ W (256 x 256), full-f32 WMMA 16x16x4.
// One 16x16 C tile per wave per nt; K stepped by 4.
// A frag (16x4 f32): lane m=lane&15, k = kbase + (lane<16 ? 0 : 2) + elem.
// B frag (4x16 f32): lane n=lane&15, same k split (mirror of A layout).
// ---------------------------------------------------------------------------
__global__ __launch_bounds__(256) void xw_gemm_kernel(const float* __restrict__ X,
                                                      const float* __restrict__ W,
                                                      float* __restrict__ XW) {
    const int m0   = blockIdx.x * 16;        // 1024 blocks cover 16384 rows
    const int wave = threadIdx.x >> 5;       // 8 waves
    const int lane = threadIdx.x & 31;
    const int half = lane >> 4;              // 0: K pair {0,1}, 1: K pair {2,3}
    const int l16  = lane & 15;
    const int row  = m0 + l16;

    for (int nt = 0; nt < 2; ++nt) {
        const int n0 = (wave * 2 + nt) * 16; // 8 waves x 2 tiles = N 0..255
        v8f acc = {};
        for (int k = 0; k < D_; k += 4) {
            const int ka = k + half * 2;
            v2f a;
            a.x = X[row * D_ + ka];
            a.y = X[row * D_ + ka + 1];
            v2f bf;
            bf.x = W[(ka    ) * OUT_ + n0 + l16];
            bf.y = W[(ka + 1) * OUT_ + n0 + l16];
            acc = __builtin_amdgcn_wmma_f32_16x16x4_f32(
                false, a, false, bf, (short)0, acc, false, false);
        }
        // C/D layout: VGPR j -> M=j (lanes 0-15) / M=j+8 (lanes 16-31), N=lane&15
        #pragma unroll
        for (int j = 0; j < 8; ++j) {
            const int r = m0 + j + half * 8;
            XW[r * OUT_ + n0 + l16] = acc[j];
        }
    }
}

// ---------------------------------------------------------------------------
// Phase 1b: S[row][r] = dot(X[row], Watt[r]) for r=0..3. One wave per row.
// ---------------------------------------------------------------------------
__global__ __launch_bounds__(256) void s_kernel(const float* __restrict__ X,
                                                const float* __restrict__ Watt,
                                                float* __restrict__ S) {
    const int wave = threadIdx.x >> 5;
    const int lane = threadIdx.x & 31;
    const int row  = blockIdx.x * 8 + wave;  // 2048 blocks x 8 waves = 16384 rows
    float a0 = 0.f, a1 = 0.f, a2 = 0.f, a3 = 0.f;
    for (int i = 0; i < D_; i += 32) {
        const float x = X[row * D_ + i + lane];
        a0 += x * Watt[0 * D_ + i + lane];
        a1 += x * Watt[1 * D_ + i + lane];
        a2 += x * Watt[2 * D_ + i + lane];
        a3 += x * Watt[3 * D_ + i + lane];
    }
    #pragma unroll
    for (int off = 16; off > 0; off >>= 1) {
        a0 += __shfl_xor(a0, off, 32);
        a1 += __shfl_xor(a1, off, 32);
        a2 += __shfl_xor(a2, off, 32);
        a3 += __shfl_xor(a3, off, 32);
    }
    if (lane == 0) {
        S[row * 4 + 0] = a0;
        S[row * 4 + 1] = a1;
        S[row * 4 + 2] = a2;
        S[row * 4 + 3] = a3;
    }
}

// ---------------------------------------------------------------------------
// Phase 2: sequential scan. One block per batch (32 blocks x 256 threads).
// Carry m (64x64) in LDS, hprev in a register, U column in 128 v2f VGPRs/lane.
// Global inputs for step t+1 are register-pipelined during step t so the
// load-wait lands after the compute, not before it.
// ---------------------------------------------------------------------------
__global__ __launch_bounds__(256) void scan_kernel(
    const float* __restrict__ XW, const float* __restrict__ S,
    const int*   __restrict__ Mp, const int*   __restrict__ Ei,
    const int*   __restrict__ Eo, const int*   __restrict__ Ri,
    const int*   __restrict__ Ro, const float* __restrict__ U,
    const float* __restrict__ bias,
    float* __restrict__ outs, float* __restrict__ mems, float* __restrict__ aggs) {

    const int b = blockIdx.x;
    const int e = threadIdx.x;

    __shared__ float m_lds[C_ * RD_];     // 16 KB carry
    __shared__ float am_lds[C_];
    __shared__ float alpha_lds[C_];
    __shared__ __align__(16) float prev_lds[OUT_];
    __shared__ float hnew_lds[OUT_];
    __shared__ int   ri_s[C_], ro_s[C_], eo_s[C_];
    __shared__ float inv_sum_s;

    // U column e resident in registers as float2 pairs (-> v_pk_fma_f32 GEMV).
    v2f ucol2[D_ / 2];
    #pragma unroll
    for (int d = 0; d < D_ / 2; ++d) {
        ucol2[d].x = U[(2 * d    ) * OUT_ + e];
        ucol2[d].y = U[(2 * d + 1) * OUT_ + e];
    }
    const float be = bias[e];

    float hprev = 0.0f;
    #pragma unroll
    for (int i = 0; i < 16; ++i) m_lds[e + i * 256] = 0.0f;
    __syncthreads();

    const int er = e >> 6;   // this thread's r for prev[]
    const int ed = e & 63;   // this thread's d for prev[]

    // ---- pipeline prologue: load step 0 inputs into registers ----
    int ri_c = 0, ro_c = 0, eo_c = 0, ei_c = 0, mt_c = 0;
    v4f  s4_c = {};
    float xe_c = 0.f;
    {
        const int rowbt = b * T_;
        const int cbase = rowbt * C_;
        if (e < C_) {
            ri_c = Ri[cbase + e];
            ro_c = Ro[cbase + e];
            eo_c = Eo[cbase + e];
            ei_c = Ei[cbase + e];
        }
        s4_c = *(const v4f*)&S[rowbt * 4];     // all 4 r-scores, 16B load
        xe_c = XW[rowbt * OUT_ + e];
        mt_c = Mp[rowbt];
    }

    for (int t = 0; t < T_; ++t) {
        const int rowbt = b * T_ + t;

        // ---- issue step t+1 loads now; waits resolve after this step ----
        int ri_n = 0, ro_n = 0, eo_n = 0, ei_n = 0, mt_n = 0;
        v4f  s4_n = {};
        float xe_n = 0.f;
        if (t + 1 < T_) {
            const int rowbt1 = rowbt + 1;
            const int cbase1 = rowbt1 * C_;
            if (e < C_) {
                ri_n = Ri[cbase1 + e];
                ro_n = Ro[cbase1 + e];
                eo_n = Eo[cbase1 + e];
                ei_n = Ei[cbase1 + e];
            }
            s4_n = *(const v4f*)&S[rowbt1 * 4];
            xe_n = XW[rowbt1 * OUT_ + e];
            mt_n = Mp[rowbt1];
        }

        // --- masked attention numerators (all operands already in regs) ---
        if (e < C_) {
            ri_s[e] = ri_c;
            ro_s[e] = ro_c;
            eo_s[e] = eo_c;
            const float sv = s4_c[ri_c];
            // EPS = 1e-100 underflows to 0.0f in f32, matching jax f32 semantics
            am_lds[e] = __expf(sv) * (float)ei_c + (float)1e-100;
        }
        __syncthreads();

        // --- softmax denominator: 64 -> 32 then wave32 xor-reduce ---
        if (e < 32) {
            float v = am_lds[e] + am_lds[e + 32];
            #pragma unroll
            for (int off = 16; off > 0; off >>= 1) v += __shfl_xor(v, off, 32);
            if (e == 0) inv_sum_s = 1.0f / v;
        }
        __syncthreads();
        if (e < C_) alpha_lds[e] = am_lds[e] * inv_sum_s;
        __syncthreads();

        // --- agg_out[r] ---
        if (e < R_) {
            float a = 0.f;
            for (int c = 0; c < C_; ++c)
                if (ri_s[c] == e) a += alpha_lds[c];
            aggs[rowbt * R_ + e] = a;
        }

        // --- prev[r*64+d] = sum_c alpha[c]*(ri[c]==r)*m[c][d] ---
        {
            float acc = 0.f;
            for (int c = 0; c < C_; ++c)
                if (ri_s[c] == er) acc += alpha_lds[c] * m_lds[c * RD_ + ed];
            prev_lds[e] = acc;
        }
        __syncthreads();

        // --- h2h[e] = prev @ U : packed f32 FMAs, b64 LDS broadcasts ---
        v2f acc2 = {0.f, 0.f};
        #pragma unroll
        for (int d = 0; d < D_ / 2; ++d) {
            const v2f p = *(const v2f*)&prev_lds[2 * d];
            acc2 = acc2 + p * ucol2[d];
        }
        const float h2h = acc2.x + acc2.y;

        // Wstk/Ustk replicate W/U, so xr=xz=xh and hr=hz=hh => rg == z.
        const float pre  = xe_c + h2h + be;
        const float z    = 1.0f / (1.0f + __expf(-pre));
        const float ht   = tanhf(xe_c + z * h2h + be);
        const float pv   = prev_lds[e];
        const float hnew = (1.0f - z) * pv + z * ht;
        const float hout = mt_c ? hnew : hprev;
        hprev = hout;
        hnew_lds[e] = hnew;
        outs[rowbt * OUT_ + e] = hout;
        __syncthreads();

        // --- memory update: mout[c][d] = (m&eo[c]) ? hnew[ro[c]*64+d] : m[c][d]
        float* memout = &mems[(size_t)rowbt * (C_ * RD_)];
        #pragma unroll
        for (int i = 0; i < 16; ++i) {
            const int idx = e + i * 256;
            const int c = idx >> 6, d = idx & 63;
            float mo = m_lds[idx];
            if (mt_c && eo_s[c]) mo = hnew_lds[ro_s[c] * RD_ + d];
            m_lds[idx] = mo;
            memout[idx] = mo;
        }
        __syncthreads();

        // ---- rotate pipeline registers ----
        ri_c = ri_n; ro_c = ro_n; eo_c = eo_n; ei_c = ei_n;
        s4_c = s4_n; xe_c = xe_n; mt_c = mt_n;
    }
}

// ---------------------------------------------------------------------------
extern "C" void kernel_launch(void* const* d_in, const int* in_sizes, int n_in,
                              void* d_out, int out_size, void* d_ws, size_t ws_size,
                              hipStream_t stream) {
    const float* X    = (const float*)d_in[0];
    const int*   M    = (const int*)  d_in[1];
    const int*   Ei   = (const int*)  d_in[2];
    const int*   Eo   = (const int*)  d_in[3];
    const int*   Ri   = (const int*)  d_in[4];
    const int*   Ro   = (const int*)  d_in[5];
    const float* W    = (const float*)d_in[6];
    const float* U    = (const float*)d_in[7];
    const float* bias = (const float*)d_in[8];
    const float* Watt = (const float*)d_in[9];

    float* XW = (float*)d_ws;                         // 16 MB
    float* S  = XW + (size_t)B_ * T_ * OUT_;          // 256 KB

    float* outs = (float*)d_out;                      // (32,512,256)
    float* mems = outs + (size_t)B_ * T_ * OUT_;      // (32,512,64,64)
    float* aggs = mems + (size_t)B_ * T_ * C_ * RD_;  // (32,512,4)

    hipLaunchKernelGGL(xw_gemm_kernel, dim3((B_ * T_) / 16), dim3(256), 0, stream,
                       X, W, XW);
    hipLaunchKernelGGL(s_kernel, dim3((B_ * T_) / 8), dim3(256), 0, stream,
                       X, Watt, S);
    hipLaunchKernelGGL(scan_kernel, dim3(B_), dim3(256), 0, stream,
                       XW, S, M, Ei, Eo, Ri, Ro, U, bias, outs, mems, aggs);
}